// NGPRadianceField_mygrid_2D3D_87359634801057
// MI455X (gfx1250) — compile-verified
//
#include <hip/hip_runtime.h>
#include <stdint.h>

// NGP hash-grid encode for MI455X (gfx1250), compile-only tuned.
//
// Roofline: out = 96 MB (HBM-bound floor ~4.1us @ 23.3 TB/s); param table
// 32.6 MB fits in 192 MB L2 -> gathers are L2-resident. WMMA is inapplicable
// (per-point gathered 8x2 reductions, no shared operand matrix). CDNA5 paths
// used instead: async global->LDS staging of the two smallest levels
// (ASYNCcnt + s_wait_asynccnt), ds_load_b64 LDS gathers, non-temporal (NT)
// stores to keep the table hot in L2, wave32-coalesced point assignment.

typedef float v4f __attribute__((ext_vector_type(4)));

#define BLOCK 256
#define PPT   4           // points per thread -> 1024 blocks for N = 2^20

// Levels 0+1: 4096 + 12168 float2 entries = 16264 entries = 130112 bytes,
// contiguous at the start of the parameter table.
#define STAGE_ENTRIES 16264
#define STAGE_CHUNKS  8132   // 16-byte chunks (130112 / 16)
#define L1_BASE       4096   // level-1 entry offset inside the staged region

__global__ __launch_bounds__(BLOCK)
void ngp_grid_encode_kernel(const float* __restrict__ x,
                            const float* __restrict__ params,
                            float* __restrict__ out, int N)
{
    constexpr int      RESV[12] = {16, 23, 32, 46, 64, 92, 128, 184, 256, 368, 512, 736};
    constexpr unsigned OFF[12]  = {0u, 4096u, 16264u, 49032u, 146368u, 408512u,
                                   932800u, 1457088u, 1981376u, 2505664u,
                                   3029952u, 3554240u};
    constexpr int      NDENSE = 5;                 // levels 0..4 dense, 5..11 hashed
    constexpr unsigned HMASK  = (1u << 19) - 1u;   // hashed size = 2^19 (power of 2)
    constexpr unsigned P1 = 2654435761u, P2 = 805459861u;

    __shared__ float2 lds_tab[STAGE_ENTRIES];      // 130112 B: levels 0+1

    const int tid = threadIdx.x;

    // ---- Async-stage levels 0 & 1 (contiguous 130112 B) into LDS ----------
    {
        unsigned  lbase = (unsigned)(uintptr_t)(void*)&lds_tab[0]; // low 32b = LDS addr
        uint64_t  gbase = (uint64_t)(uintptr_t)params;             // level-0 offset = 0
        for (int k = tid; k < STAGE_CHUNKS; k += BLOCK) {
            unsigned laddr = lbase + (unsigned)k * 16u;
            uint64_t gaddr = gbase + (uint64_t)k * 16u;
            asm volatile("global_load_async_to_lds_b128 %0, %1, off"
                         :: "v"(laddr), "v"(gaddr) : "memory");
        }
        asm volatile("s_wait_asynccnt 0" ::: "memory");
    }
    __syncthreads();   // LDS writes of all waves complete before gathers

    const int base = blockIdx.x * (BLOCK * PPT);

    for (int k = 0; k < PPT; ++k) {
        const int n = base + tid + k * BLOCK;      // consecutive lanes -> consecutive n
        if (n >= N) break;

        const float px = __builtin_nontemporal_load(x + 3 * (size_t)n + 0);
        const float py = __builtin_nontemporal_load(x + 3 * (size_t)n + 1);
        const float pz = __builtin_nontemporal_load(x + 3 * (size_t)n + 2);

        float acc[24];

        #pragma unroll
        for (int l = 0; l < 12; ++l) {
            const int   R   = RESV[l];
            const float Rm1 = (float)(R - 1);
            const float Rm2 = (float)(R - 2);

            const float sx = px * Rm1, sy = py * Rm1, sz = pz * Rm1;
            const float fx = fminf(fmaxf(floorf(sx), 0.0f), Rm2);
            const float fy = fminf(fmaxf(floorf(sy), 0.0f), Rm2);
            const float fz = fminf(fmaxf(floorf(sz), 0.0f), Rm2);
            const float wx = sx - fx, wy = sy - fy, wz = sz - fz;
            const unsigned x0 = (unsigned)fx, y0 = (unsigned)fy, z0 = (unsigned)fz;

            unsigned idx[8];
            if (l < NDENSE) {
                const unsigned uR  = (unsigned)R;
                const unsigned uR2 = (unsigned)(R * R);
                const unsigned b   = x0 + y0 * uR + z0 * uR2;
                idx[0] = b;           idx[1] = b + 1u;
                idx[2] = b + uR;      idx[3] = b + uR + 1u;
                idx[4] = b + uR2;     idx[5] = b + uR2 + 1u;
                idx[6] = b + uR2 + uR; idx[7] = b + uR2 + uR + 1u;
            } else {
                const unsigned hx0 = x0,        hx1 = x0 + 1u;
                const unsigned hy0 = y0 * P1,   hy1 = hy0 + P1;   // uint32 wraparound
                const unsigned hz0 = z0 * P2,   hz1 = hz0 + P2;
                idx[0] = (hx0 ^ hy0 ^ hz0) & HMASK;
                idx[1] = (hx1 ^ hy0 ^ hz0) & HMASK;
                idx[2] = (hx0 ^ hy1 ^ hz0) & HMASK;
                idx[3] = (hx1 ^ hy1 ^ hz0) & HMASK;
                idx[4] = (hx0 ^ hy0 ^ hz1) & HMASK;
                idx[5] = (hx1 ^ hy0 ^ hz1) & HMASK;
                idx[6] = (hx0 ^ hy1 ^ hz1) & HMASK;
                idx[7] = (hx1 ^ hy1 ^ hz1) & HMASK;
            }

            float2 f[8];
            if (l == 0) {                      // LDS gather (ds_load_b64)
                #pragma unroll
                for (int c = 0; c < 8; ++c) f[c] = lds_tab[idx[c]];
            } else if (l == 1) {
                #pragma unroll
                for (int c = 0; c < 8; ++c) f[c] = lds_tab[L1_BASE + idx[c]];
            } else {                           // L2-resident global gather (b64)
                const float2* __restrict__ tab = (const float2*)params + OFF[l];
                #pragma unroll
                for (int c = 0; c < 8; ++c) f[c] = tab[idx[c]];
            }

            const float ux = 1.0f - wx, uy = 1.0f - wy, uz = 1.0f - wz;
            const float w0 = ux * uy * uz, w1 = wx * uy * uz;
            const float w2 = ux * wy * uz, w3 = wx * wy * uz;
            const float w4 = ux * uy * wz, w5 = wx * uy * wz;
            const float w6 = ux * wy * wz, w7 = wx * wy * wz;

            acc[2 * l + 0] = w0 * f[0].x + w1 * f[1].x + w2 * f[2].x + w3 * f[3].x
                           + w4 * f[4].x + w5 * f[5].x + w6 * f[6].x + w7 * f[7].x;
            acc[2 * l + 1] = w0 * f[0].y + w1 * f[1].y + w2 * f[2].y + w3 * f[3].y
                           + w4 * f[4].y + w5 * f[5].y + w6 * f[6].y + w7 * f[7].y;
        }

        // 24 floats per point = 6 x b128 non-temporal stores (don't pollute L2).
        v4f* o4 = (v4f*)(out + (size_t)n * 24);
        #pragma unroll
        for (int j = 0; j < 6; ++j) {
            v4f v = { acc[4 * j + 0], acc[4 * j + 1], acc[4 * j + 2], acc[4 * j + 3] };
            __builtin_nontemporal_store(v, o4 + j);
        }
    }
}

extern "C" void kernel_launch(void* const* d_in, const int* in_sizes, int n_in,
                              void* d_out, int out_size, void* d_ws, size_t ws_size,
                              hipStream_t stream) {
    const float* x      = (const float*)d_in[0];   // [N,3] f32
    const float* params = (const float*)d_in[1];   // [4078528,2] f32
    float*       out    = (float*)d_out;           // [N,24] f32
    const int N = in_sizes[0] / 3;

    const int blocks = (N + BLOCK * PPT - 1) / (BLOCK * PPT);
    ngp_grid_encode_kernel<<<blocks, BLOCK, 0, stream>>>(x, params, out, N);
}